// BatchedVariableSelectionNetwork_54176717471839
// MI455X (gfx1250) — compile-verified
//
#include <hip/hip_runtime.h>

typedef float v2f __attribute__((ext_vector_type(2)));
typedef float v8f __attribute__((ext_vector_type(8)));

#define NROWS 4096   // B*T
#define NV    204    // variables
#define NH    128    // hidden

__device__ __forceinline__ float elu_f(float x)    { return x > 0.f ? x : expm1f(x); }
__device__ __forceinline__ float sigmoid_f(float x){ return 1.f / (1.f + __expf(-x)); }

// ---------------------------------------------------------------------------
// Pass 1: weight network -> wts (N, V).  Small GEMMs, one block per row.
// ---------------------------------------------------------------------------
__global__ __launch_bounds__(128) void vsn_weight_kernel(
    const float* __restrict__ x,
    const float* __restrict__ wfc1_w, const float* __restrict__ wfc1_b,
    const float* __restrict__ wfc2_w, const float* __restrict__ wfc2_b,
    const float* __restrict__ wglu_w, const float* __restrict__ wglu_b,
    const float* __restrict__ wln_g,  const float* __restrict__ wln_b,
    float* __restrict__ wts_out)
{
    __shared__ float xs_s[NV];
    __shared__ float a1_s[NH];
    __shared__ float a2_s[NH];
    __shared__ float g_s[2 * NV];
    __shared__ float t_s[NV];
    __shared__ float rs[128];
    __shared__ float rq[128];

    const int n   = blockIdx.x;
    const int tid = threadIdx.x;

    for (int i = tid; i < NV; i += 128) xs_s[i] = x[n * NV + i];
    __syncthreads();

    // a1 = elu(xs @ wfc1_w + b)
    float acc = wfc1_b[tid];
    for (int v = 0; v < NV; ++v) acc = fmaf(xs_s[v], wfc1_w[v * NH + tid], acc);
    a1_s[tid] = elu_f(acc);
    __syncthreads();

    // a2 = a1 @ wfc2_w + b
    acc = wfc2_b[tid];
    for (int k = 0; k < NH; ++k) acc = fmaf(a1_s[k], wfc2_w[k * NH + tid], acc);
    a2_s[tid] = acc;
    __syncthreads();

    // g = a2 @ wglu_w + b    (2V = 408 outputs)
    for (int j = tid; j < 2 * NV; j += 128) {
        float a = wglu_b[j];
        for (int k = 0; k < NH; ++k) a = fmaf(a2_s[k], wglu_w[k * (2 * NV) + j], a);
        g_s[j] = a;
    }
    __syncthreads();

    // s = xs + glu(g); mean/var partials
    float ps = 0.f, pq = 0.f;
    for (int v = tid; v < NV; v += 128) {
        float hg = g_s[v] * sigmoid_f(g_s[NV + v]);
        float s  = xs_s[v] + hg;
        t_s[v] = s;
        ps += s; pq += s * s;
    }
    rs[tid] = ps; rq[tid] = pq; __syncthreads();
    for (int off = 64; off > 0; off >>= 1) {
        if (tid < off) { rs[tid] += rs[tid + off]; rq[tid] += rq[tid + off]; }
        __syncthreads();
    }
    const float mu  = rs[0] * (1.f / NV);
    const float var = rq[0] * (1.f / NV) - mu * mu;
    const float inv = rsqrtf(var + 1e-5f);
    __syncthreads();

    // t = LN(s); max-reduce for softmax
    float pm = -3.402823466e38f;
    for (int v = tid; v < NV; v += 128) {
        float t = fmaf((t_s[v] - mu) * inv, wln_g[v], wln_b[v]);
        t_s[v] = t;
        pm = fmaxf(pm, t);
    }
    rs[tid] = pm; __syncthreads();
    for (int off = 64; off > 0; off >>= 1) {
        if (tid < off) rs[tid] = fmaxf(rs[tid], rs[tid + off]);
        __syncthreads();
    }
    const float tmax = rs[0];
    __syncthreads();

    float pe = 0.f;
    for (int v = tid; v < NV; v += 128) {
        float e = __expf(t_s[v] - tmax);
        t_s[v] = e;
        pe += e;
    }
    rq[tid] = pe; __syncthreads();
    for (int off = 64; off > 0; off >>= 1) {
        if (tid < off) rq[tid] += rq[tid + off];
        __syncthreads();
    }
    const float rsum = 1.f / rq[0];
    for (int v = tid; v < NV; v += 128) wts_out[n * NV + v] = t_s[v] * rsum;
}

// ---------------------------------------------------------------------------
// Pass 2: per-variable GRN with f32 WMMA (16x16x4), fused LN + weighted sum.
// One block = 16 rows of N; 8 waves; loops all 204 variables; out written once.
// ---------------------------------------------------------------------------
__global__ __launch_bounds__(256) void vsn_var_kernel(
    const float* __restrict__ x,
    const float* __restrict__ fc1_w, const float* __restrict__ fc1_b,
    const float* __restrict__ fc2_w, const float* __restrict__ fc2_b,
    const float* __restrict__ glu_w, const float* __restrict__ glu_b,
    const float* __restrict__ skip_w, const float* __restrict__ skip_b,
    const float* __restrict__ ln_g,  const float* __restrict__ ln_b,
    const float* __restrict__ wts,   float* __restrict__ out)
{
    __shared__ float xs_s [16 * NV];
    __shared__ float wts_s[16 * NV];
    __shared__ float h1s  [16 * NH];
    __shared__ float h2s  [16 * NH];
    __shared__ float ys   [16 * NH];

    const int tid = threadIdx.x;
    const int n0  = blockIdx.x * 16;

    for (int i = tid; i < 16 * NV; i += 256) {
        int r = i / NV, c = i % NV;
        xs_s[i]  = x  [(n0 + r) * NV + c];
        wts_s[i] = wts[(n0 + r) * NV + c];
    }

    const int wave  = tid >> 5;
    const int lane  = tid & 31;
    const int l15   = lane & 15;
    const int khalf = (lane >> 4) << 1;   // A/B fragment k shift: 0 or 2
    const int mbase = (lane >> 4) << 3;   // C/D row base: 0 or 8

    // LN-affine params + output accumulators in reduction layout:
    // thread t owns row (t>>4), cols [(t&15)*8, +8)
    const int rrow = tid >> 4;
    const int rcol = (tid & 15) << 3;
    float lng[8], lnb[8], oacc[8];
    for (int j = 0; j < 8; ++j) { lng[j] = ln_g[rcol + j]; lnb[j] = ln_b[rcol + j]; oacc[j] = 0.f; }
    __syncthreads();

    for (int v = 0; v < NV; ++v) {
        // ---- Phase A: h1 = elu(xs[:,v] (outer) fc1_w[v] + fc1_b[v]), 16x128
        const float* f1w = fc1_w + v * NH;
        const float* f1b = fc1_b + v * NH;
        for (int i = tid; i < 16 * NH; i += 256) {
            int r = i >> 7, h = i & 127;
            float t = fmaf(xs_s[r * NV + v], f1w[h], f1b[h]);
            h1s[i] = elu_f(t);
        }
        __syncthreads();

        // ---- Phase B: h2 = h1 @ fc2_w[v] + fc2_b[v]; wave owns 16 out cols
        const float* W2 = fc2_w + (size_t)v * NH * NH;
        const int col1 = wave * 16 + l15;
        v8f acc1;
        { float b = fc2_b[v * NH + col1]; for (int r = 0; r < 8; ++r) acc1[r] = b; }
        #pragma unroll 4
        for (int k = 0; k < NH; k += 4) {
            int kk = k + khalf;
            v2f a = *(const v2f*)&h1s[l15 * NH + kk];           // A[m=l15][kk,kk+1]
            v2f b; b[0] = W2[kk * NH + col1];                    // B[kk][col]
                   b[1] = W2[(kk + 1) * NH + col1];              // B[kk+1][col]
            acc1 = __builtin_amdgcn_wmma_f32_16x16x4_f32(false, a, false, b,
                                                         (short)0, acc1, false, false);
        }
        for (int r = 0; r < 8; ++r) h2s[(mbase + r) * NH + col1] = acc1[r];
        __syncthreads();

        // ---- Phase C: glu = h2 @ glu_w[v]^T + glu_b[v]; wave owns matching
        //      fc_out cols [16w,+16) and gate cols [128+16w,+16)
        const float* GW = glu_w + (size_t)v * 2 * NH * NH;
        const int gcol = wave * 16 + l15;
        v8f accF, accG;
        { float bF = glu_b[v * 2 * NH + gcol];
          float bG = glu_b[v * 2 * NH + NH + gcol];
          for (int r = 0; r < 8; ++r) { accF[r] = bF; accG[r] = bG; } }
        const float* rowF = GW + (size_t)gcol * NH;        // glu_w[v][gcol][:]
        const float* rowG = GW + (size_t)(NH + gcol) * NH; // glu_w[v][128+gcol][:]
        #pragma unroll 4
        for (int k = 0; k < NH; k += 4) {
            int kk = k + khalf;
            v2f a  = *(const v2f*)&h2s[l15 * NH + kk];
            v2f bF = *(const v2f*)&rowF[kk];               // k-contiguous (vgh layout)
            v2f bG = *(const v2f*)&rowG[kk];
            accF = __builtin_amdgcn_wmma_f32_16x16x4_f32(false, a, false, bF,
                                                         (short)0, accF, false, false);
            accG = __builtin_amdgcn_wmma_f32_16x16x4_f32(false, a, false, bG,
                                                         (short)0, accG, false, false);
        }

        // ---- Phase D1: GLU + skip, write y tile
        const float sw = skip_w[v * NH + gcol];
        const float sb = skip_b[v * NH + gcol];
        for (int r = 0; r < 8; ++r) {
            int m = mbase + r;
            float hv = accF[r] * sigmoid_f(accG[r]);
            float yv = fmaf(xs_s[m * NV + v], sw, sb) + hv;
            ys[m * NH + gcol] = yv;
        }
        __syncthreads();

        // ---- Phase D2: row-wise LN stats (shfl width 16) + weighted accumulate
        float yv[8], s1 = 0.f, s2 = 0.f;
        for (int j = 0; j < 8; ++j) {
            yv[j] = ys[rrow * NH + rcol + j];
            s1 += yv[j]; s2 += yv[j] * yv[j];
        }
        for (int off = 8; off > 0; off >>= 1) {
            s1 += __shfl_xor(s1, off, 16);
            s2 += __shfl_xor(s2, off, 16);
        }
        const float mu   = s1 * (1.f / NH);
        const float varr = s2 * (1.f / NH) - mu * mu;
        const float inv  = rsqrtf(varr + 1e-5f);
        const float wt   = wts_s[rrow * NV + v];
        for (int j = 0; j < 8; ++j) {
            float t = fmaf((yv[j] - mu) * inv, lng[j], lnb[j]);
            oacc[j] = fmaf(wt, t, oacc[j]);
        }
        // next iteration's barriers (post-A, post-B) already order ys reuse
    }

    for (int j = 0; j < 8; ++j)
        out[(n0 + rrow) * NH + rcol + j] = oacc[j];
}

// ---------------------------------------------------------------------------
extern "C" void kernel_launch(void* const* d_in, const int* in_sizes, int n_in,
                              void* d_out, int out_size, void* d_ws, size_t ws_size,
                              hipStream_t stream)
{
    const float* x      = (const float*)d_in[0];
    const float* fc1_w  = (const float*)d_in[1];
    const float* fc1_b  = (const float*)d_in[2];
    const float* fc2_w  = (const float*)d_in[3];
    const float* fc2_b  = (const float*)d_in[4];
    const float* glu_w  = (const float*)d_in[5];
    const float* glu_b  = (const float*)d_in[6];
    const float* skip_w = (const float*)d_in[7];
    const float* skip_b = (const float*)d_in[8];
    const float* ln_g   = (const float*)d_in[9];
    const float* ln_b   = (const float*)d_in[10];
    const float* wfc1_w = (const float*)d_in[11];
    const float* wfc1_b = (const float*)d_in[12];
    const float* wfc2_w = (const float*)d_in[13];
    const float* wfc2_b = (const float*)d_in[14];
    const float* wglu_w = (const float*)d_in[15];
    const float* wglu_b = (const float*)d_in[16];
    const float* wln_g  = (const float*)d_in[17];
    const float* wln_b  = (const float*)d_in[18];

    float* out_main = (float*)d_out;               // (B,T,H) = N*128
    float* wts_out  = out_main + NROWS * NH;       // (B,T,V) = N*204

    vsn_weight_kernel<<<dim3(NROWS), dim3(128), 0, stream>>>(
        x, wfc1_w, wfc1_b, wfc2_w, wfc2_b, wglu_w, wglu_b, wln_g, wln_b, wts_out);

    vsn_var_kernel<<<dim3(NROWS / 16), dim3(256), 0, stream>>>(
        x, fc1_w, fc1_b, fc2_w, fc2_b, glu_w, glu_b, skip_w, skip_b,
        ln_g, ln_b, wts_out, out_main);
}